// SemanticMultiGroupConv_53901839565172
// MI455X (gfx1250) — compile-verified
//
#include <hip/hip_runtime.h>
#include <hip/hip_bf16.h>

typedef __attribute__((ext_vector_type(2))) float v2f;
typedef __attribute__((ext_vector_type(8))) float v8f;

#define NG   16      // groups / independent convs
#define NB   2       // batch
#define NC   256     // channels
#define CG   16      // channels per group
#define NH   24
#define NW   24
#define HW   576     // 24*24
#define PH   26      // padded spatial
#define PHW  676     // 26*26
#define NT   72      // (NB*HW)/16
#define AC   6400    // 25*NC affinity channels
#define BN_EPS 1e-5f

// ---------------------------------------------------------------------------
// Prep A: zero-padded input, channel-PAIR interleaved:
//   xpad2[(chpair*PHW + s)*2 + (ch&1)],  chpair = (bb*NC + c) >> 1.
// Gives the conv kernel one aligned b64 load per B fragment.
// Total elements: NB*NC*PHW = 346112.
// ---------------------------------------------------------------------------
__global__ __launch_bounds__(256) void pad_input_pairs(
    const float* __restrict__ x, float* __restrict__ xpad2)
{
  const int t    = blockIdx.x * 256 + threadIdx.x;   // < 346112
  const int e    = t & 1;
  const int rest = t >> 1;                           // chpair*PHW + s
  const int s    = rest % PHW;
  const int cp   = rest / PHW;                       // 0..255
  const int ch   = cp * 2 + e;                       // bb*NC + c, 0..511
  const int y    = s / PH;
  const int xx   = s - y * PH;
  float v = 0.0f;
  if ((unsigned)(y - 1) < (unsigned)NH && (unsigned)(xx - 1) < (unsigned)NW)
    v = x[(size_t)ch * HW + (y - 1) * NW + (xx - 1)];
  xpad2[t] = v;
}

// ---------------------------------------------------------------------------
// Prep B: weight re-layout. K order kidx = (ky*3+kx)*16 + ic.
// Wt[pair][kp][m][2], k = kp*2 + e -> one b64 A-fragment load per step.
// ---------------------------------------------------------------------------
__global__ __launch_bounds__(256) void prep_weights(
    const float* __restrict__ W1, float* __restrict__ Wt)
{
  const int t    = blockIdx.x * 256 + threadIdx.x; // < 589824
  const int e    = t & 1;
  const int m    = (t >> 1) & 15;
  const int kp   = (t >> 5) % 72;
  const int pair = t / 2304;
  const int k    = kp * 2 + e;                     // kidx = r*16 + ic
  const int r    = k >> 4;                         // ky*3 + kx
  const int ic   = k & 15;
  const int g    = pair & 15;
  const int i    = pair >> 4;
  Wt[t] = W1[(size_t)(i * NC + g * CG + m) * 144 + ic * 9 + r];
}

// ---------------------------------------------------------------------------
// Stage 1: 16 grouped 3x3 convs + BN + ReLU -> h1i (pair-interleaved):
//   h1i[(((i*NB+bb)*128 + chpair)*HW + rem)*2 + (ch&1)]
// One wave per 16x16 tile, 36 x V_WMMA_F32_16X16X4_F32; per step:
// one b64 A load + one b64 B load, all immediate offsets, branchless.
// ---------------------------------------------------------------------------
__global__ __launch_bounds__(256) void conv3_bn_relu(
    const float* __restrict__ xpad2, const float* __restrict__ Wt,
    const float* __restrict__ g1, const float* __restrict__ b1,
    const float* __restrict__ m1, const float* __restrict__ v1,
    float* __restrict__ h1i)
{
  const int lane = threadIdx.x & 31;
  const int wave = threadIdx.x >> 5;
  const int tile = blockIdx.x * 8 + wave;     // 0 .. 18431
  const int nt   = tile % NT;
  const int pair = tile / NT;                 // 0 .. 255
  const int g    = pair & 15;
  const int i    = pair >> 4;

  const int half = lane >> 4;
  const int mr   = lane & 15;

  const int n   = nt * 16 + mr;               // 16-wide tile stays in one batch
  const int bb  = n / HW;
  const int rem = n - bb * HW;
  const int yy  = rem / NW;
  const int xx  = rem - yy * NW;

  // A base: Wt[pair][kp = r*8+cc*2+half][m][2].
  const float* wt = Wt + (size_t)pair * 2304 + half * 32 + mr * 2;
  // B base: pair-interleaved padded image; lane's ic = cc*4 + half*2 (even),
  // so its (ic, ic+1) pair is one contiguous b64: pair index g*8 + cc*2 + half.
  const float* pb = xpad2
      + 2 * ((size_t)(bb * 128 + g * 8 + half) * PHW + yy * PH + xx);

  v8f acc = {};
  #pragma unroll
  for (int r = 0; r < 9; ++r) {               // r = ky*3 + kx
    const int ky = r / 3;
    const int kx = r - ky * 3;
    #pragma unroll
    for (int cc = 0; cc < 4; ++cc) {          // ic block of 4
      v2f a  = *(const v2f*)(wt + (r * 8 + cc * 2) * 32);
      v2f bf = *(const v2f*)(pb + (cc * 2 * PHW + ky * PH + kx) * 2);
      acc = __builtin_amdgcn_wmma_f32_16x16x4_f32(false, a, false, bf,
                                                  (short)0, acc, false, false);
    }
  }

  // Fused BN + ReLU; adjacent output channels (v, v+1) are one b64 store.
  const int chbase = g * CG + half * 8;       // even
  #pragma unroll
  for (int v = 0; v < 8; v += 2) {
    const int p0 = i * NC + chbase + v;
    const float sc0 = g1[p0]     * rsqrtf(v1[p0]     + BN_EPS);
    const float sc1 = g1[p0 + 1] * rsqrtf(v1[p0 + 1] + BN_EPS);
    v2f st;
    st.x = fmaxf((acc[v]     - m1[p0])     * sc0 + b1[p0],     0.0f);
    st.y = fmaxf((acc[v + 1] - m1[p0 + 1]) * sc1 + b1[p0 + 1], 0.0f);
    const int cp = (chbase + v) >> 1;
    *(v2f*)(h1i + ((size_t)((i * NB + bb) * 128 + cp) * HW + rem) * 2) = st;
  }
}

// ---------------------------------------------------------------------------
// Stage 2: grouped 1x1 conv + BN + ReLU + interior spatial sum.
// Per wave: fixed (i, gc, mtile); weights preloaded, sweep 72 N-tiles,
// 4 WMMAs each (1 b64 B load per WMMA); shfl row-reduce; direct stores.
// ---------------------------------------------------------------------------
__global__ __launch_bounds__(256) void conv1_bn_relu_pool(
    const float* __restrict__ h1i, const float* __restrict__ W2,
    const float* __restrict__ g2, const float* __restrict__ b2,
    const float* __restrict__ m2, const float* __restrict__ v2,
    float* __restrict__ poolsum)
{
  const int lane = threadIdx.x & 31;
  const int wave = threadIdx.x >> 5;
  int tile = blockIdx.x * 8 + wave;           // 0 .. 6399
  const int mt = tile % 25; tile /= 25;
  const int gc = tile & 15;
  const int i  = tile >> 4;

  const int half = lane >> 4;
  const int mr   = lane & 15;

  // Preload 16x16 weight block as 4 A-fragments (contiguous pairs).
  const float* wbase = W2 + (size_t)(i * AC + gc * 400 + mt * 16 + mr) * CG
                          + half * 2;
  v2f aw[4];
  #pragma unroll
  for (int kk = 0; kk < 4; ++kk)
    aw[kk] = *(const v2f*)(wbase + kk * 4);

  // Hoist BN params for this wave's output channels (8 per half).
  float sc[8], mm[8], bbp[8];
  #pragma unroll
  for (int v = 0; v < 8; ++v) {
    const int pidx = i * AC + gc * 400 + mt * 16 + v + half * 8;
    sc[v]  = g2[pidx] * rsqrtf(v2[pidx] + BN_EPS);
    mm[v]  = m2[pidx];
    bbp[v] = b2[pidx];
  }

  for (int bb = 0; bb < NB; ++bb) {
    // Lane's K channels gc*16 + kk*4 + half*2 + {0,1} -> pair gc*8 + kk*2 + half.
    const float* hbase = h1i
        + 2 * ((size_t)((i * NB + bb) * 128 + gc * 8 + half) * HW + mr);
    float rs[8];
    #pragma unroll
    for (int v = 0; v < 8; ++v) rs[v] = 0.0f;

    for (int nt = 0; nt < 36; ++nt) {
      const float* hp = hbase + nt * 32;      // rem advances 16 -> 32 floats
      v8f acc = {};
      #pragma unroll
      for (int kk = 0; kk < 4; ++kk) {
        v2f bf = *(const v2f*)(hp + kk * 2 * HW * 2);
        acc = __builtin_amdgcn_wmma_f32_16x16x4_f32(false, aw[kk], false, bf,
                                                    (short)0, acc, false, false);
      }
      #pragma unroll
      for (int v = 0; v < 8; ++v)
        rs[v] += fmaxf((acc[v] - mm[v]) * sc[v] + bbp[v], 0.0f);
    }

    #pragma unroll
    for (int v = 0; v < 8; ++v) {
      float val = rs[v];
      for (int m = 1; m < 16; m <<= 1) val += __shfl_xor(val, m, 32);
      if (mr == 0) {
        const int och = gc * 400 + mt * 16 + v + half * 8;
        poolsum[(size_t)(i * NB + bb) * AC + och] = val;
      }
    }
  }
}

// ---------------------------------------------------------------------------
// Stage 3: pooled = (interior_sum + 100*relu(b2 - m2*scale2)) / 676;
// aff_i[i,b,l] = dot(xv[i,b,i,:], xv[i,b,l,:]) / 16 over m in [0,400).
// ---------------------------------------------------------------------------
__global__ __launch_bounds__(256) void affinity(
    const float* __restrict__ poolsum,
    const float* __restrict__ g2, const float* __restrict__ b2,
    const float* __restrict__ m2, const float* __restrict__ v2,
    float* __restrict__ aff)
{
  const int ib = blockIdx.x;                  // (i,b), 0..31
  const int i  = ib >> 1;
  const int t  = threadIdx.x;
  const int l  = t & 15;
  const int part = t >> 4;

  const float* ps  = poolsum + (size_t)ib * AC;
  const int    iAC = i * AC;

  float dot = 0.0f;
  for (int m = part; m < 400; m += 16) {
    const int ot = i * 400 + m;
    const int ol = l * 400 + m;
    const float sct = g2[iAC + ot] * rsqrtf(v2[iAC + ot] + BN_EPS);
    const float scl = g2[iAC + ol] * rsqrtf(v2[iAC + ol] + BN_EPS);
    const float th = ps[ot] + 100.0f * fmaxf(b2[iAC + ot] - m2[iAC + ot] * sct, 0.0f);
    const float ph = ps[ol] + 100.0f * fmaxf(b2[iAC + ol] - m2[iAC + ol] * scl, 0.0f);
    dot += th * ph;
  }

  __shared__ float red[256];
  red[t] = dot;
  __syncthreads();
  #pragma unroll
  for (int s = 128; s >= 16; s >>= 1) {
    if (t < s) red[t] += red[t + s];
    __syncthreads();
  }
  if (t < 16) {
    const float inv = 1.0f / (676.0f * 676.0f * 16.0f);
    aff[ib * 16 + t] = red[t] * inv;
  }
}

// ---------------------------------------------------------------------------
// Stage 4: out[p, (i*2+q)*16+c, y, x] = sum_k aff[i,p,k] * h1[i,q,k*16+c,y,x]
// h1i base: channel k*16+c -> pair k*8 + (c>>1), element (c&1); k-stride is
// 8 pairs * HW * 2 = 16*HW floats (same as non-interleaved layout).
// ---------------------------------------------------------------------------
__global__ __launch_bounds__(256) void combine(
    const float* __restrict__ h1i, const float* __restrict__ aff,
    float* __restrict__ out)
{
  const int t  = blockIdx.x * 256 + threadIdx.x;  // < 589824
  const int s  = t % HW;
  const int r  = t / HW;
  const int ch = r % 512;
  const int p  = r / 512;
  const int c  = ch & 15;
  const int q  = (ch >> 4) & 1;
  const int i  = ch >> 5;

  const float* hb = h1i
      + ((size_t)((i * NB + q) * 128 + (c >> 1)) * HW + s) * 2 + (c & 1);
  const float* ab = aff + (i * NB + p) * NG;
  float sum = 0.0f;
  #pragma unroll
  for (int k = 0; k < NG; ++k)
    sum += ab[k] * hb[(size_t)k * CG * HW];
  out[t] = sum;
}

// ---------------------------------------------------------------------------
extern "C" void kernel_launch(void* const* d_in, const int* in_sizes, int n_in,
                              void* d_out, int out_size, void* d_ws, size_t ws_size,
                              hipStream_t stream)
{
  const float* x  = (const float*)d_in[0];
  const float* W1 = (const float*)d_in[1];
  const float* g1 = (const float*)d_in[2];
  const float* b1 = (const float*)d_in[3];
  const float* m1 = (const float*)d_in[4];
  const float* v1 = (const float*)d_in[5];
  const float* W2 = (const float*)d_in[6];
  const float* g2 = (const float*)d_in[7];
  const float* b2 = (const float*)d_in[8];
  const float* m2 = (const float*)d_in[9];
  const float* v2 = (const float*)d_in[10];
  float* out = (float*)d_out;

  float* h1i     = (float*)d_ws;                         // 4,718,592 f
  float* poolsum = h1i + (size_t)NG * NB * NC * HW;      //   204,800 f
  float* affw    = poolsum + (size_t)NG * NB * AC;       //       512 f
  float* xpad2   = affw + 512;                           //   346,112 f
  float* Wt      = xpad2 + (size_t)NB * NC * PHW;        //   589,824 f

  pad_input_pairs   <<<1352, 256, 0, stream>>>(x, xpad2);       // 346112/256
  prep_weights      <<<2304, 256, 0, stream>>>(W1, Wt);         // 589824/256
  conv3_bn_relu     <<<2304, 256, 0, stream>>>(xpad2, Wt, g1, b1, m1, v1, h1i);
  conv1_bn_relu_pool<<< 800, 256, 0, stream>>>(h1i, W2, g2, b2, m2, v2, poolsum);
  affinity          <<<  32, 256, 0, stream>>>(poolsum, g2, b2, m2, v2, affw);
  combine           <<<2304, 256, 0, stream>>>(h1i, affw, out);
}